// ContrastiveLoss_85950885528563
// MI455X (gfx1250) — compile-verified
//
#include <hip/hip_runtime.h>
#include <stdint.h>

namespace {

constexpr int kD     = 256;       // feature dim
constexpr int kNeg   = 32;        // negatives per query
constexpr int kPairs = kNeg + 1;  // 33 neighbor rows per query
constexpr int kWaves = 4;         // waves (queries) per block; 33KB LDS each

typedef uint32_t v4u __attribute__((ext_vector_type(4)));
typedef uint32_t v8u __attribute__((ext_vector_type(8)));
typedef float    v2f __attribute__((ext_vector_type(2)));
typedef float    v8f __attribute__((ext_vector_type(8)));

// TDM waits must also be compiler memory barriers so the LDS reads that
// consume DMA'd data cannot be hoisted above them.
#define WAIT_TENSORCNT(n) asm volatile("s_wait_tensorcnt " #n ::: "memory")

// One TDM gather descriptor: fetch `nidx` rows (16-bit row indices packed two
// per dword in p0..p3) of a row-major [nrows_tensor x 256] fp32 tensor into
// LDS at byte offset `lds_addr`.  D# layout per CDNA5 ISA.
__device__ __forceinline__ void tdm_gather(uint32_t lds_addr, const float* tensor,
                                           uint32_t nrows_tensor, uint32_t nidx,
                                           uint32_t p0, uint32_t p1,
                                           uint32_t p2, uint32_t p3) {
  const uint64_t ga = (uint64_t)(uintptr_t)tensor;
  // group0: count=1, gather_mode=1 (bit31), 16-bit indices (bit30=0),
  //         lds_addr, global_addr[56:0], type=2 (bits 127:126)
  v4u g0 = { 0x80000001u,
             lds_addr,
             (uint32_t)ga,
             ((uint32_t)(ga >> 32) & 0x01FFFFFFu) | 0x80000000u };
  // group1: wg_mask=0, data_size=2 (4B), no barrier/iterate/pad,
  //         tensor_dim0=256, tensor_dim1=nrows_tensor, tile_dim0=256,
  //         tile_dim1=nidx, tensor_dim0_stride=256
  v8u g1 = { 0x00020000u,
             ((uint32_t)kD & 0xFFFFu) << 16,
             (nrows_tensor & 0xFFFFu) << 16,
             ((uint32_t)kD << 16) | ((nrows_tensor >> 16) & 0xFFFFu),
             nidx,
             (uint32_t)kD,
             0u, 0u };
  // groups 2/3: up to 16 u16 row indices (we use at most 8 per descriptor)
  v4u g2 = { p0, p1, p2, p3 };
  v4u g3 = { 0u, 0u, 0u, 0u };
  asm volatile("tensor_load_to_lds %0, %1, %2, %3"
               :: "s"(g0), "s"(g1), "s"(g2), "s"(g3) : "memory");
}

// Exact 32-lane sum via two V_WMMA_F32_16X16X4_F32 with B = all-ones.
// W1: A = {v,0} -> D[m][n] = v[m] + v[m+16] (documented A layout; B layout-free).
// Fold: per-lane sum of the 8 D VGPRs = half-sums (documented C/D layout).
// W2: same trick -> every element of D2 equals the full 32-lane sum, so the
// final read is position-independent.  EXEC is all-1s here as WMMA requires.
__device__ __forceinline__ float wave_sum_wmma(float v) {
  const v2f ones = { 1.0f, 1.0f };
  v8f c = {};
  v2f a = { v, 0.0f };
  v8f d1 = __builtin_amdgcn_wmma_f32_16x16x4_f32(
      false, a, false, ones, (short)0, c, false, false);
  float e = d1[0] + d1[1] + d1[2] + d1[3] + d1[4] + d1[5] + d1[6] + d1[7];
  v2f a2 = { e, 0.0f };
  v8f d2 = __builtin_amdgcn_wmma_f32_16x16x4_f32(
      false, a2, false, ones, (short)0, c, false, false);
  return d2[0];
}

__global__ __launch_bounds__(kWaves * 32) void
contrastive_loss_main(const float* __restrict__ feats,
                      const int*   __restrict__ negs,
                      float* __restrict__ block_sums,
                      int b) {
  __shared__ float stage[kWaves][kPairs][kD];   // 132 KB staging
  __shared__ float wave_loss[kWaves];

  const int lane = threadIdx.x & 31;                                 // wave32
  const int wid  = __builtin_amdgcn_readfirstlane(threadIdx.x >> 5); // uniform
  const int q    = blockIdx.x * kWaves + wid;                        // uniform

  // Query row in registers: lane covers bytes [16*lane,16*lane+16) per 512B half.
  const float* fq = feats + (size_t)q * kD;
  const float4 a0 = *(const float4*)(fq + 4 * lane);
  const float4 a1 = *(const float4*)(fq + kD / 2 + 4 * lane);

  // Neighbor row indices (uniform scalar loads; all < 2b = 65536 -> fit u16).
  uint32_t r[kPairs + 1];
  r[0] = (uint32_t)(b + q);                       // positive pair
  const int* nq = negs + (size_t)q * kNeg;
#pragma unroll
  for (int j = 0; j < kNeg; ++j) r[1 + j] = (uint32_t)nq[j];
  r[kPairs] = 0u;                                 // pad for pair packing

  auto pk = [&](int i) {                          // pack two u16 indices
    return ((r[i + 1] & 0xFFFFu) << 16) | (r[i] & 0xFFFFu);
  };

  const uint32_t twob = 2u * (uint32_t)b;
  const uint32_t lds0 = (uint32_t)(uintptr_t)&stage[wid][0][0];
  constexpr uint32_t kChunkBytes = 8u * kD * 4u;  // 8 rows = 8KB

  // Issue 5 gather descriptors (8+8+8+8+1 rows = 33); TDM completes in order.
  tdm_gather(lds0 + 0 * kChunkBytes, feats, twob, 8, pk(0),  pk(2),  pk(4),  pk(6));
  tdm_gather(lds0 + 1 * kChunkBytes, feats, twob, 8, pk(8),  pk(10), pk(12), pk(14));
  tdm_gather(lds0 + 2 * kChunkBytes, feats, twob, 8, pk(16), pk(18), pk(20), pk(22));
  tdm_gather(lds0 + 3 * kChunkBytes, feats, twob, 8, pk(24), pk(26), pk(28), pk(30));
  tdm_gather(lds0 + 4 * kChunkBytes, feats, twob, 1, r[32] & 0xFFFFu, 0u, 0u, 0u);

  float pos = 0.0f, negsum = 0.0f;

  auto consume = [&](int pair) {
    const float4* row = (const float4*)&stage[wid][pair][0];
    const float4 b0 = row[lane];            // first 512B of the row
    const float4 b1 = row[kD / 8 + lane];   // second 512B
    float t, acc = 0.0f;
    t = a0.x - b0.x; acc = fmaf(t, t, acc);
    t = a0.y - b0.y; acc = fmaf(t, t, acc);
    t = a0.z - b0.z; acc = fmaf(t, t, acc);
    t = a0.w - b0.w; acc = fmaf(t, t, acc);
    t = a1.x - b1.x; acc = fmaf(t, t, acc);
    t = a1.y - b1.y; acc = fmaf(t, t, acc);
    t = a1.z - b1.z; acc = fmaf(t, t, acc);
    t = a1.w - b1.w; acc = fmaf(t, t, acc);
    // wave32 cross-lane tree reduction of the 8-dim partials
    acc += __shfl_xor(acc, 16, 32);
    acc += __shfl_xor(acc, 8, 32);
    acc += __shfl_xor(acc, 4, 32);
    acc += __shfl_xor(acc, 2, 32);
    acc += __shfl_xor(acc, 1, 32);
    float p = 1.0f / (1.0f + acc);          // EPS = 1.0
    p = fminf(fmaxf(p, 1.0e-4f), 1.0f);     // clip(probits, 1e-4, 1.0)
    if (pair == 0) pos = p; else negsum += p;
  };

  // Consume chunk k once TENSORcnt <= (4-k): overlap math with remaining DMA.
  WAIT_TENSORCNT(4);
  for (int p = 0; p < 8; ++p) consume(p);
  WAIT_TENSORCNT(3);
  for (int p = 8; p < 16; ++p) consume(p);
  WAIT_TENSORCNT(2);
  for (int p = 16; p < 24; ++p) consume(p);
  WAIT_TENSORCNT(1);
  for (int p = 24; p < 32; ++p) consume(p);
  WAIT_TENSORCNT(0);
  consume(32);

  // TEMPERATURE/BASE_TEMPERATURE == 1.0
  const float loss = __logf(negsum) - __logf(pos);  // -(log pos - log negsum)

  if (lane == 0) wave_loss[wid] = loss;
  __syncthreads();
  if (threadIdx.x == 0) {
    float s = 0.0f;
#pragma unroll
    for (int w = 0; w < kWaves; ++w) s += wave_loss[w];
    block_sums[blockIdx.x] = s;
  }
}

__global__ __launch_bounds__(256) void
reduce_block_sums(const float* __restrict__ block_sums, int n,
                  float* __restrict__ out) {
  __shared__ float wtot[8];
  const int lane = threadIdx.x & 31;
  const int wid  = threadIdx.x >> 5;

  float s = 0.0f;
  for (int i = threadIdx.x; i < n; i += 256) s += block_sums[i];

  // 32-lane sum on the matrix pipe (all lanes active -> EXEC all-1s).
  const float wsum = wave_sum_wmma(s);

  if (lane == 0) wtot[wid] = wsum;
  __syncthreads();
  if (threadIdx.x == 0) {
    float t = 0.0f;
#pragma unroll
    for (int w = 0; w < 8; ++w) t += wtot[w];
    out[0] = t;
  }
}

} // namespace

extern "C" void kernel_launch(void* const* d_in, const int* in_sizes, int n_in,
                              void* d_out, int out_size, void* d_ws, size_t ws_size,
                              hipStream_t stream) {
  const float* feats = (const float*)d_in[0];   // (2b, 256) fp32
  const int*   negs  = (const int*)d_in[1];     // (b, 32) int32
  float* out = (float*)d_out;                   // scalar
  float* block_sums = (float*)d_ws;             // b/kWaves floats of scratch

  const int b      = in_sizes[1] / kNeg;        // 32768
  const int blocks = b / kWaves;                // 8192

  contrastive_loss_main<<<dim3(blocks), dim3(kWaves * 32), 0, stream>>>(
      feats, negs, block_sums, b);
  reduce_block_sums<<<dim3(1), dim3(256), 0, stream>>>(block_sums, blocks, out);
}